// CorrectSplineLinear_60129542144395
// MI455X (gfx1250) — compile-verified
//
#include <hip/hip_runtime.h>

typedef __attribute__((ext_vector_type(2))) float v2f;
typedef __attribute__((ext_vector_type(4))) float v4f;
typedef __attribute__((ext_vector_type(8))) float v8f;

// Problem constants: O=256, I=512, K=3, C=128, B=128
// y[c,b] = sum_f E[c,f] * x[b,f]    E:(128,512) x:(128,512) -> y:(128,128)
__global__ __launch_bounds__(32)
void spline_gemm_y(const float* __restrict__ E, const float* __restrict__ x,
                   float* __restrict__ y) {
    const int lane = threadIdx.x & 31;
    const int half = lane >> 4;        // 0: K pair {kk,kk+1}, 1: {kk+2,kk+3}
    const int l    = lane & 15;
    const int rowA = blockIdx.x * 16 + l;   // c (A matrix row, M)
    const int rowB = blockIdx.y * 16 + l;   // b (B matrix col, N) -> row of x
    v8f acc = {};
    #pragma unroll 4
    for (int kk = 0; kk < 512; kk += 4) {
        const int kb = kk + half * 2;
        v2f a = *(const v2f*)(E + rowA * 512 + kb);   // A[M=c][K=f]
        v2f b = *(const v2f*)(x + rowB * 512 + kb);   // B[K=f][N=b] = x[b][f]
        acc = __builtin_amdgcn_wmma_f32_16x16x4_f32(
            false, a, false, b, (short)0, acc, false, false);
    }
    const int col   = blockIdx.y * 16 + l;
    const int rbase = blockIdx.x * 16 + half * 8;
    #pragma unroll
    for (int r = 0; r < 8; ++r)
        y[(rbase + r) * 128 + col] = acc[r];
}

// Z[m,b] = sum_c CV[m,c] * y[c,b]   CV:(1024,128) y:(128,128) -> Z:(1024,128)
// (control_values (O,K+1,C) flattened row-major is exactly CV (1024,128))
__global__ __launch_bounds__(32)
void spline_gemm_z(const float* __restrict__ CV, const float* __restrict__ y,
                   float* __restrict__ Z) {
    const int lane = threadIdx.x & 31;
    const int half = lane >> 4;
    const int l    = lane & 15;
    const int rowA = blockIdx.x * 16 + l;   // m
    const int colB = blockIdx.y * 16 + l;   // b
    v8f acc = {};
    #pragma unroll 4
    for (int kk = 0; kk < 128; kk += 4) {
        const int kb = kk + half * 2;
        v2f a = *(const v2f*)(CV + rowA * 128 + kb);  // A[M=m][K=c]
        v2f b;
        b.x = y[kb * 128 + colB];                     // B[K=c][N=b] = y[c][b]
        b.y = y[(kb + 1) * 128 + colB];
        acc = __builtin_amdgcn_wmma_f32_16x16x4_f32(
            false, a, false, b, (short)0, acc, false, false);
    }
    const int col   = blockIdx.y * 16 + l;
    const int rbase = blockIdx.x * 16 + half * 8;
    #pragma unroll
    for (int r = 0; r < 8; ++r)
        Z[(rbase + r) * 128 + col] = acc[r];
}

// out[o,b,t] = (1-tl)*Z[o*4+j, b] + tl*Z[o*4+j+1, b]
// t in [0,512): ts = t*3/511; j = clip(floor(ts),0,2); tl = ts - j
// 67 MB streaming write, HBM-bound (~2.9us at 23.3 TB/s).
// One float4 (4 consecutive t) per thread -> global_store_b128.
__global__ __launch_bounds__(256)
void spline_interp_out(const float* __restrict__ Z, float* __restrict__ out) {
    const int idx4 = blockIdx.x * 256 + threadIdx.x;  // 4,194,304 groups of 4
    const int t4 = idx4 & 127;                        // 128 float4 per (o,b) row
    const int b  = (idx4 >> 7) & 127;
    const int o  = idx4 >> 14;
    const float* zbase = Z + (o * 4) * 128 + b;       // L2-resident (512 KB)
    v4f v;
    #pragma unroll
    for (int i = 0; i < 4; ++i) {
        const int t = t4 * 4 + i;
        float ts = (float)t * (3.0f / 511.0f);
        int j = (int)ts;
        if (j > 2) j = 2;
        float tl = ts - (float)j;
        float z0 = zbase[j * 128];                    // broadcast loads in-wave
        float z1 = zbase[(j + 1) * 128];
        v[i] = z0 + tl * (z1 - z0);
    }
    *(v4f*)(out + idx4 * 4) = v;                      // 16B store, coalesced
}

extern "C" void kernel_launch(void* const* d_in, const int* in_sizes, int n_in,
                              void* d_out, int out_size, void* d_ws, size_t ws_size,
                              hipStream_t stream) {
    const float* x  = (const float*)d_in[0];          // (128, 512)
    // d_in[1] = control_points: only its shape (K+1=4) matters; values unused
    const float* cv = (const float*)d_in[2];          // (256, 4, 128)
    const float* E  = (const float*)d_in[3];          // (128, 512)
    float* out = (float*)d_out;                       // (256, 128, 512)

    float* y = (float*)d_ws;                                        // 64 KB
    float* Z = (float*)((char*)d_ws + 128 * 128 * sizeof(float));   // 512 KB

    spline_gemm_y<<<dim3(8, 8), 32, 0, stream>>>(E, x, y);
    spline_gemm_z<<<dim3(64, 8), 32, 0, stream>>>(cv, y, Z);
    spline_interp_out<<<16384, 256, 0, stream>>>(Z, out);
}